// GNN_59863254171698
// MI455X (gfx1250) — compile-verified
//
#include <hip/hip_runtime.h>

#define N_NODES  100000
#define N_EDGES  600000
#define N_GRAPHS 512
#define DIM      128
#define N_LAYERS 3
#define N_CLASSES 2
#define BN_EPS   1e-5f

// LDS row pitch (floats): 130 keeps ds_load_b64 8B-aligned (even) and spreads
// the 16 same-offset lane reads across banks ((row*130 + k) % 64 varies w/ row).
#define LPITCH 130

typedef float v2f __attribute__((ext_vector_type(2)));
typedef float v8f __attribute__((ext_vector_type(8)));

// ---------------------------------------------------------------------------
// Elementwise helpers
// ---------------------------------------------------------------------------
__global__ __launch_bounds__(256) void gin_copy_f4(const float4* __restrict__ src,
                                                   float4* __restrict__ dst, int n4) {
  int i = blockIdx.x * blockDim.x + threadIdx.x;
  if (i < n4) dst[i] = src[i];
}

__global__ __launch_bounds__(256) void gin_zero_f(float* __restrict__ p, int n) {
  int i = blockIdx.x * blockDim.x + threadIdx.x;
  if (i < n) p[i] = 0.f;
}

// ---------------------------------------------------------------------------
// Edge scatter: agg[dst] += h[src]   (one wave per edge, lanes cover 128 feats
// as float4 chunks -> coalesced 512B gather + contiguous f32 atomics)
// ---------------------------------------------------------------------------
__global__ __launch_bounds__(256) void gin_scatter_edges(const float* __restrict__ h,
                                                         const int* __restrict__ src,
                                                         const int* __restrict__ dst,
                                                         float* __restrict__ agg) {
  long long t = (long long)blockIdx.x * blockDim.x + threadIdx.x;
  int e = (int)(t >> 5);
  if (e >= N_EDGES) return;
  int c0 = ((int)(t & 31)) * 4;
  const float4 v = *(const float4*)(h + (size_t)src[e] * DIM + c0);
  float* p = agg + (size_t)dst[e] * DIM + c0;
  atomicAdd(p + 0, v.x);
  atomicAdd(p + 1, v.y);
  atomicAdd(p + 2, v.z);
  atomicAdd(p + 3, v.w);
}

// ---------------------------------------------------------------------------
// GEMM + bias with native f32 WMMA: C[M x 128] = A[M x 128] @ W[128 x 128] + b
// Block (256 thr = 8 waves) computes a 32x128 output slab.
//  - A tile (32x128) and W (128x128, TRANSPOSED) staged in padded LDS.
//  - Each wave: two 16x16 tiles (shared B fragment), K-loop of 32 steps,
//    3x ds_load_b64 + 2x V_WMMA_F32_16X16X4_F32 per step, fully unrolled.
// Requires M % 32 == 0 (N_NODES = 100000 = 3125 * 32).
// ---------------------------------------------------------------------------
__global__ __launch_bounds__(256) void gin_gemm_bias_wmma(const float* __restrict__ A,
                                                          const float* __restrict__ W,
                                                          const float* __restrict__ bias,
                                                          float* __restrict__ C, int M) {
  __shared__ __align__(16) float sA[32 * LPITCH];   // A rows m0..m0+31
  __shared__ __align__(16) float sW[DIM * LPITCH];  // sW[n][k] = W[k][n]

  const int tid = threadIdx.x;
  const int m0  = blockIdx.x * 32;

  // Stage A tile: 32 rows x 128 cols = 1024 float4, coalesced global reads.
  #pragma unroll
  for (int i = tid; i < 32 * 32; i += 256) {
    const int r  = i >> 5;
    const int c4 = (i & 31) * 4;
    const float4 v = *(const float4*)(A + (size_t)(m0 + r) * DIM + c4);
    float* p = sA + r * LPITCH + c4;
    p[0] = v.x; p[1] = v.y; p[2] = v.z; p[3] = v.w;
  }
  // Stage W transposed: coalesced global reads, scattered LDS writes.
  #pragma unroll
  for (int i = tid; i < DIM * 32; i += 256) {
    const int k  = i >> 5;
    const int n4 = (i & 31) * 4;
    const float4 v = *(const float4*)(W + (size_t)k * DIM + n4);
    sW[(n4 + 0) * LPITCH + k] = v.x;
    sW[(n4 + 1) * LPITCH + k] = v.y;
    sW[(n4 + 2) * LPITCH + k] = v.z;
    sW[(n4 + 3) * LPITCH + k] = v.w;
  }
  __syncthreads();

  const int wave = tid >> 5;
  const int lane = tid & 31;
  const int half = lane >> 4;   // selects K pair {0,1} vs {2,3} of each step
  const int l16  = lane & 15;
  const int n0   = wave * 16;

  const float* aRow0 = sA + l16 * LPITCH;          // A frag rows, tile 0
  const float* aRow1 = sA + (16 + l16) * LPITCH;   // A frag rows, tile 1
  const float* bRow  = sW + (n0 + l16) * LPITCH;   // B frag column (transposed)

  v8f c0 = {};
  v8f c1 = {};
  #pragma unroll
  for (int k = 0; k < DIM; k += 4) {
    const int ka = k + 2 * half;
    const v2f a0 = *(const v2f*)(aRow0 + ka);
    const v2f a1 = *(const v2f*)(aRow1 + ka);
    const v2f b  = *(const v2f*)(bRow + ka);
    c0 = __builtin_amdgcn_wmma_f32_16x16x4_f32(false, a0, false, b,
                                               (short)0, c0, false, false);
    c1 = __builtin_amdgcn_wmma_f32_16x16x4_f32(false, a1, false, b,
                                               (short)0, c1, false, false);
  }

  const int col = n0 + l16;
  const float bv = bias[col];
  #pragma unroll
  for (int r = 0; r < 8; ++r) {
    // C/D layout: lanes 0-15 hold M=r, lanes 16-31 hold M=8+r
    const int row = m0 + half * 8 + r;
    C[(size_t)row * DIM + col]        = c0[r] + bv;
    C[(size_t)(row + 16) * DIM + col] = c1[r] + bv;
  }
}

// ---------------------------------------------------------------------------
// Per-column sum / sum-of-squares (thread = column, grid-stride over rows)
// ---------------------------------------------------------------------------
__global__ __launch_bounds__(DIM) void gin_col_stats(const float* __restrict__ t,
                                                     float* __restrict__ s,
                                                     float* __restrict__ sq, int M) {
  const int c = threadIdx.x;
  float acc = 0.f, acc2 = 0.f;
  for (int r = blockIdx.x; r < M; r += gridDim.x) {
    const float v = t[(size_t)r * DIM + c];
    acc += v;
    acc2 += v * v;
  }
  atomicAdd(&s[c], acc);
  atomicAdd(&sq[c], acc2);
}

// ---------------------------------------------------------------------------
// Fold BN1 into the second GEMM:  BN(x)@W2 + b2  ==  x@Wf + bf
//   Wf[j][k] = scale[j]*W2[j][k],  bf[k] = b2[k] + sum_j shift[j]*W2[j][k]
//   scale[j] = rsqrt(v+eps)*g1[j],  shift[j] = bt1[j] - m[j]*scale[j]
// ---------------------------------------------------------------------------
__global__ __launch_bounds__(DIM) void gin_fold_bn(const float* __restrict__ W2,
                                                   const float* __restrict__ b2,
                                                   const float* __restrict__ g1,
                                                   const float* __restrict__ bt1,
                                                   const float* __restrict__ s,
                                                   const float* __restrict__ sq,
                                                   float* __restrict__ Wf,
                                                   float* __restrict__ bf, float invN) {
  __shared__ float scale[DIM];
  __shared__ float shift[DIM];
  const int t = threadIdx.x;
  const float m  = s[t] * invN;
  const float v  = sq[t] * invN - m * m;
  const float sc = rsqrtf(v + BN_EPS) * g1[t];
  scale[t] = sc;
  shift[t] = bt1[t] - m * sc;
  __syncthreads();
  float acc = 0.f;
  #pragma unroll 8
  for (int j = 0; j < DIM; ++j) {
    const float w = W2[j * DIM + t];
    Wf[j * DIM + t] = scale[j] * w;
    acc += shift[j] * w;
  }
  bf[t] = b2[t] + acc;
}

// ---------------------------------------------------------------------------
// Composed BN2 + outer-BN: mean(BN2 out)=bt2 exactly, var = c2^2*v2 with
// c2 = rsqrt(v2+eps)*g2, so the outer BN stats come for free:
//   h = relu( t2*alpha + beta ),  alpha = c2*rsqrt(c2^2*v2+eps)*go,
//   beta = bo - m2*alpha            (bt2 cancels)
// ---------------------------------------------------------------------------
__global__ __launch_bounds__(DIM) void gin_outer_prep(const float* __restrict__ s,
                                                      const float* __restrict__ sq,
                                                      const float* __restrict__ g2,
                                                      const float* __restrict__ go,
                                                      const float* __restrict__ bo,
                                                      float* __restrict__ alpha,
                                                      float* __restrict__ beta,
                                                      float invN) {
  const int t = threadIdx.x;
  const float m2 = s[t] * invN;
  const float v2 = sq[t] * invN - m2 * m2;
  const float c2 = rsqrtf(v2 + BN_EPS) * g2[t];
  const float vo = c2 * c2 * v2;
  const float a  = c2 * rsqrtf(vo + BN_EPS) * go[t];
  alpha[t] = a;
  beta[t]  = bo[t] - m2 * a;
}

__global__ __launch_bounds__(256) void gin_apply_relu(const float* __restrict__ t2,
                                                      const float* __restrict__ alpha,
                                                      const float* __restrict__ beta,
                                                      float* __restrict__ h, int n) {
  int i = blockIdx.x * blockDim.x + threadIdx.x;
  if (i < n) {
    const int c = i & (DIM - 1);
    const float v = t2[i] * alpha[c] + beta[c];
    h[i] = v > 0.f ? v : 0.f;
  }
}

// ---------------------------------------------------------------------------
// Pooling
// ---------------------------------------------------------------------------
__global__ __launch_bounds__(256) void gin_pool_scatter(const float* __restrict__ h,
                                                        const int* __restrict__ batch,
                                                        float* __restrict__ gsum) {
  long long t = (long long)blockIdx.x * blockDim.x + threadIdx.x;
  int node = (int)(t >> 5);
  if (node >= N_NODES) return;
  int c0 = ((int)(t & 31)) * 4;
  const float4 v = *(const float4*)(h + (size_t)node * DIM + c0);
  float* p = gsum + (size_t)batch[node] * DIM + c0;
  atomicAdd(p + 0, v.x);
  atomicAdd(p + 1, v.y);
  atomicAdd(p + 2, v.z);
  atomicAdd(p + 3, v.w);
}

__global__ __launch_bounds__(256) void gin_count_nodes(const int* __restrict__ batch,
                                                       float* __restrict__ counts) {
  int i = blockIdx.x * blockDim.x + threadIdx.x;
  if (i < N_NODES) atomicAdd(&counts[batch[i]], 1.0f);
}

__global__ __launch_bounds__(256) void gin_pool_finalize(float* __restrict__ gsum,
                                                         const float* __restrict__ counts) {
  int i = blockIdx.x * blockDim.x + threadIdx.x;
  if (i < N_GRAPHS * DIM) {
    const float cnt = counts[i >> 7];
    gsum[i] = gsum[i] / (cnt > 1.0f ? cnt : 1.0f);
  }
}

__global__ __launch_bounds__(256) void gin_classifier(const float* __restrict__ ge,
                                                      const float* __restrict__ fcW,
                                                      const float* __restrict__ fcb,
                                                      float* __restrict__ out) {
  int i = blockIdx.x * blockDim.x + threadIdx.x;
  if (i >= N_GRAPHS * N_CLASSES) return;
  const int g = i >> 1, k = i & 1;
  float acc = fcb[k];
  #pragma unroll 8
  for (int c = 0; c < DIM; ++c) acc += ge[g * DIM + c] * fcW[c * N_CLASSES + k];
  out[i] = acc;
}

// ---------------------------------------------------------------------------
// Launcher
// ---------------------------------------------------------------------------
extern "C" void kernel_launch(void* const* d_in, const int* in_sizes, int n_in,
                              void* d_out, int out_size, void* d_ws, size_t ws_size,
                              hipStream_t stream) {
  const float* x    = (const float*)d_in[0];
  const int*   ei   = (const int*)d_in[1];
  const int*   bat  = (const int*)d_in[2];
  const float* W1   = (const float*)d_in[3];
  const float* b1   = (const float*)d_in[4];
  const float* g1   = (const float*)d_in[5];
  const float* bt1  = (const float*)d_in[6];
  const float* W2   = (const float*)d_in[7];
  const float* b2   = (const float*)d_in[8];
  const float* g2   = (const float*)d_in[9];
  const float* bt2  = (const float*)d_in[10]; (void)bt2; // cancels analytically
  const float* go   = (const float*)d_in[11];
  const float* bo   = (const float*)d_in[12];
  const float* fcW  = (const float*)d_in[13];
  const float* fcb  = (const float*)d_in[14];

  const int* esrc = ei;
  const int* edst = ei + N_EDGES;

  const size_t NF = (size_t)N_NODES * DIM;

  // Output layout: node_embeddings | graph_embedding | out
  float* h    = (float*)d_out;                 // N_NODES*DIM (evolves in place)
  float* gsum = h + NF;                        // N_GRAPHS*DIM
  float* out  = gsum + (size_t)N_GRAPHS * DIM; // N_GRAPHS*N_CLASSES

  // Workspace layout
  float* agg   = (float*)d_ws;        // N_NODES*DIM (also reused as t2)
  float* t1    = agg + NF;            // N_NODES*DIM
  float* small = t1 + NF;
  float* msum  = small;               // 128
  float* vsum  = small + DIM;         // 128
  float* alpha = small + 2 * DIM;     // 128
  float* beta  = small + 3 * DIM;     // 128
  float* Wf    = small + 4 * DIM;     // 128*128
  float* bf    = Wf + DIM * DIM;      // 128
  float* cnts  = bf + DIM;            // 512
  float* t2    = agg;                 // reuse

  const float invN = 1.0f / (float)N_NODES;

  const int n4        = (int)(NF / 4);
  const int copyGrid  = (n4 + 255) / 256;
  const int ewGrid    = (int)(((long long)N_EDGES * 32 + 255) / 256);
  const int nwGrid    = (int)(((long long)N_NODES * 32 + 255) / 256);
  const int elemGrid  = (int)((NF + 255) / 256);
  const int gemmGrid  = N_NODES / 32;  // 3125 (exact; M % 32 == 0)

  // h = x
  gin_copy_f4<<<copyGrid, 256, 0, stream>>>((const float4*)x, (float4*)h, n4);

  for (int i = 0; i < N_LAYERS; ++i) {
    const float* W1i = W1 + (size_t)i * DIM * DIM;
    const float* W2i = W2 + (size_t)i * DIM * DIM;

    // agg = h (self term), then agg += scatter of neighbors
    gin_copy_f4<<<copyGrid, 256, 0, stream>>>((const float4*)h, (float4*)agg, n4);
    gin_scatter_edges<<<ewGrid, 256, 0, stream>>>(h, esrc, edst, agg);

    // t1 = agg @ W1[i] + b1[i]
    gin_gemm_bias_wmma<<<gemmGrid, 256, 0, stream>>>(agg, W1i, b1 + i * DIM, t1, N_NODES);

    // BN1 stats, fold into W2
    gin_zero_f<<<1, 256, 0, stream>>>(msum, 2 * DIM);
    gin_col_stats<<<256, DIM, 0, stream>>>(t1, msum, vsum, N_NODES);
    gin_fold_bn<<<1, DIM, 0, stream>>>(W2i, b2 + i * DIM, g1 + i * DIM, bt1 + i * DIM,
                                       msum, vsum, Wf, bf, invN);

    // t2 = BN1(t1) @ W2[i] + b2[i]  ==  t1 @ Wf + bf
    gin_gemm_bias_wmma<<<gemmGrid, 256, 0, stream>>>(t1, Wf, bf, t2, N_NODES);

    // BN2 stats -> composed BN2+outerBN+ReLU into h
    gin_zero_f<<<1, 256, 0, stream>>>(msum, 2 * DIM);
    gin_col_stats<<<256, DIM, 0, stream>>>(t2, msum, vsum, N_NODES);
    gin_outer_prep<<<1, DIM, 0, stream>>>(msum, vsum, g2 + i * DIM, go + i * DIM,
                                          bo + i * DIM, alpha, beta, invN);
    gin_apply_relu<<<elemGrid, 256, 0, stream>>>(t2, alpha, beta, h, (int)NF);
  }

  // Mean pooling per graph
  gin_zero_f<<<(N_GRAPHS * DIM + 255) / 256, 256, 0, stream>>>(gsum, N_GRAPHS * DIM);
  gin_zero_f<<<(N_GRAPHS + 255) / 256, 256, 0, stream>>>(cnts, N_GRAPHS);
  gin_pool_scatter<<<nwGrid, 256, 0, stream>>>(h, bat, gsum);
  gin_count_nodes<<<(N_NODES + 255) / 256, 256, 0, stream>>>(bat, cnts);
  gin_pool_finalize<<<(N_GRAPHS * DIM + 255) / 256, 256, 0, stream>>>(gsum, cnts);

  // Classifier
  gin_classifier<<<(N_GRAPHS * N_CLASSES + 255) / 256, 256, 0, stream>>>(gsum, fcW, fcb, out);
}